// SpectralCoherence_5325759447204
// MI455X (gfx1250) — compile-verified
//
#include <hip/hip_runtime.h>
#include <math.h>

// CDNA5 wave32 WMMA types: f32 16x16x4 -> A/B = 2 VGPRs/lane, C/D = 8 VGPRs/lane
typedef __attribute__((ext_vector_type(2))) float v2f;
typedef __attribute__((ext_vector_type(8))) float v8f;

#define NK1 128
#define NK2 256
#define NK3 256
#define NYZ 16

__device__ __forceinline__ float trapzw(const float* x, int i, int n) {
    int ip = (i + 1 < n) ? i + 1 : n - 1;
    int im = (i - 1 > 0) ? i - 1 : 0;
    return 0.5f * (x[ip] - x[im]);
}

__global__ __launch_bounds__(512) void spectral_coherence_kernel(
    const float* __restrict__ k1_in, const float* __restrict__ dy_in,
    const float* __restrict__ dz_in, const float* __restrict__ k2_in,
    const float* __restrict__ k3_in, const float* __restrict__ Lp,
    const float* __restrict__ Mp,   const float* __restrict__ Tp,
    float* __restrict__ out)
{
    __shared__ float sT1[NK2 * NYZ];    // T1[b][z] = sum_c Phi*w3*cos(dz*k3)
    __shared__ float sT2[NK2 * NYZ];    // T2[b][z] = sum_c Phi*w3*sin(dz*k3)
    __shared__ float sTabC[NK3 * NYZ];  // stage1: w3[c]*cos(dz[z]*k3[c]) ; stage2: w2[b]*cos(dy[y]*k2[b])
    __shared__ float sTabS[NK3 * NYZ];  // stage1: w3[c]*sin(dz[z]*k3[c]) ; stage2: -w2[b]*sin(dy[y]*k2[b])
    __shared__ float sFi[16];
    __shared__ float sW2[NK2], sW3[NK3];
    __shared__ float sK2[NK2], sK3[NK3], sDy[NYZ], sDz[NYZ];

    const int a    = blockIdx.x;
    const int tid  = threadIdx.x;
    const int lane = tid & 31;
    const int wave = tid >> 5;          // 16 waves: one b-tile each
    const int row  = lane & 15;         // A: M-row / B,C: N-col within tile
    const int hi   = lane >> 4;         // 0: K=0,1 ; 1: K=2,3 (A & B layouts)

    for (int i = tid; i < NK2; i += 512) { sK2[i] = k2_in[i]; sK3[i] = k3_in[i]; }
    if (tid < NYZ) { sDy[tid] = dy_in[tid]; sDz[tid] = dz_in[tid]; }
    __syncthreads();

    // ---- Cold setup: trapezoid-weight tables + stage-1 trig tables --------
    for (int i = tid; i < NK2; i += 512) {
        sW2[i] = trapzw(sK2, i, NK2);
        sW3[i] = trapzw(sK3, i, NK3);
    }
    for (int idx = tid; idx < NK3 * NYZ; idx += 512) {
        const int   c   = idx >> 4;
        const int   z   = idx & 15;
        const float w3c = trapzw(sK3, c, NK3);
        float sv, cv;
        sincosf(sDz[z] * sK3[c], &sv, &cv);
        sTabC[idx] = w3c * cv;
        sTabS[idx] = w3c * sv;
    }
    __syncthreads();

    const float L  = Lp[0], Mg = Mp[0], Ts = Tp[0];
    const float k1 = k1_in[a];

    // Per-lane (k1,k2) invariants for Phi11
    const int   b0  = wave * 16;
    const int   b   = b0 + row;            // this lane's b-row (A matrix M)
    const float k2v = sK2[b];
    const float w2b = sW2[b];
    const float k1s = k1 * k1;
    const float k2s = k2v * k2v;
    const float s   = k1s + k2s;
    const float sqs = sqrtf(s);
    const float inv_s15 = 1.0f / (s * sqs);
    const float k2ok1   = k2v / k1;
    const float L2      = L * L;
    const float Epref   = Mg / (4.0f * 3.14159265358979323846f);

    // ---- Stage 1: contract over c with V_WMMA_F32_16X16X4_F32 -------------
    v8f   ccos = {};
    v8f   csin = {};
    float fiAcc = 0.0f;

    for (int c0 = 0; c0 < NK3; c0 += 4) {
        v2f A, Bc, Bs;
#pragma unroll
        for (int j = 0; j < 2; ++j) {
            const int   c   = c0 + 2 * hi + j;  // K index for both A and B
            const float k3v = sK3[c];
            // --- Phi11(k1, k2v, k3v), invariants hoisted ---
            const float kk   = s + k3v * k3v;
            const float beta = Ts / cbrtf(L2 * kk);            // Ts*(L*sqrt(kk))^(-2/3)
            const float k30  = k3v + beta * k1;
            const float kk0  = s + k30 * k30;
            const float k0L2 = L2 * kk0;
            // (1+k0L2)^(-17/6) via hardware log2/exp2 (base > 1, no specials)
            const float p176 = exp2f(-17.0f / 6.0f * log2f(1.0f + k0L2));
            const float E0   = Epref * (k0L2 * k0L2) * p176;
            const float C1   = beta * k1s * (kk0 - 2.0f * k30 * k30 + beta * k1 * k30)
                               / (kk * s);
            const float C2   = k2v * kk0 * inv_s15 *
                               atanf(beta * k1 * sqs / (kk0 - k30 * k1 * beta));
            const float z1   = C1 - k2ok1 * C2;
            const float P    = E0 / (kk0 * kk0) *
                               (kk0 - k1s - 2.0f * k1 * k30 * z1 + s * z1 * z1);
            A[j] = P;
            fiAcc += P * sW3[c];
            const int ti = (c << 4) | row;      // table index [c][z=row]
            Bc[j] = sTabC[ti];
            Bs[j] = sTabS[ti];
        }
        // D = A(16x4 f32) * B(4x16 f32) + C  (f32 accumulate, wave32)
        ccos = __builtin_amdgcn_wmma_f32_16x16x4_f32(false, A, false, Bc,
                                                     (short)0, ccos, false, false);
        csin = __builtin_amdgcn_wmma_f32_16x16x4_f32(false, A, false, Bs,
                                                     (short)0, csin, false, false);
    }

    // Wave-level Fi reduction (xor shuffle -> all lanes hold wave total)
    fiAcc *= w2b;
#pragma unroll
    for (int off = 16; off > 0; off >>= 1) fiAcc += __shfl_xor(fiAcc, off, 32);
    if (lane == 0) sFi[wave] = fiAcc;

    // C layout: VGPR r, lanes 0-15 -> M=r, lanes 16-31 -> M=r+8; N = lane%16
#pragma unroll
    for (int r = 0; r < 8; ++r) {
        const int brow = r + hi * 8;
        sT1[(b0 + brow) * NYZ + row] = ccos[r];
        sT2[(b0 + brow) * NYZ + row] = csin[r];
    }
    __syncthreads();

    // ---- Build stage-2 tables (overwrite sTab*): w2*cos/-sin(dy*k2) -------
    for (int idx = tid; idx < NK2 * NYZ; idx += 512) {
        const int   bb  = idx >> 4;
        const int   y   = idx & 15;
        const float w2k = sW2[bb];
        float sv, cv;
        sincosf(sDy[y] * sK2[bb], &sv, &cv);
        sTabC[idx] =  w2k * cv;
        sTabS[idx] = -w2k * sv;
    }
    float fi = 0.0f;
#pragma unroll
    for (int i = 0; i < 16; ++i) fi += sFi[i];
    __syncthreads();

    // ---- Stage 2: Re[y,z] = sum_b Cy[y,b]*T1[b,z] - Sy[y,b]*T2[b,z] -------
    // Executed redundantly by ALL waves (keeps EXEC all-1s for WMMA);
    // every wave computes and stores identical values.
    v8f cre = {};
    for (int bb = 0; bb < NK2; bb += 4) {
        v2f Ac, As, B1, B2;
#pragma unroll
        for (int j = 0; j < 2; ++j) {
            const int bk = bb + 2 * hi + j;     // K index over b
            const int ti = (bk << 4) | row;
            Ac[j] = sTabC[ti];                  // w2*cos(dy[y=row]*k2[bk])
            As[j] = sTabS[ti];                  // -w2*sin(...)
            B1[j] = sT1[ti];
            B2[j] = sT2[ti];
        }
        cre = __builtin_amdgcn_wmma_f32_16x16x4_f32(false, Ac, false, B1,
                                                    (short)0, cre, false, false);
        cre = __builtin_amdgcn_wmma_f32_16x16x4_f32(false, As, false, B2,
                                                    (short)0, cre, false, false);
    }

    const float inv = 1.0f / fi;
#pragma unroll
    for (int r = 0; r < 8; ++r) {
        const int y = r + hi * 8;
        out[a * (NYZ * NYZ) + y * NYZ + row] = cre[r] * inv;
    }
}

extern "C" void kernel_launch(void* const* d_in, const int* in_sizes, int n_in,
                              void* d_out, int out_size, void* d_ws, size_t ws_size,
                              hipStream_t stream) {
    (void)in_sizes; (void)n_in; (void)out_size; (void)d_ws; (void)ws_size;
    const float* k1 = (const float*)d_in[0];
    const float* dy = (const float*)d_in[1];
    const float* dz = (const float*)d_in[2];
    const float* k2 = (const float*)d_in[3];
    const float* k3 = (const float*)d_in[4];
    const float* Lp = (const float*)d_in[5];
    const float* Mp = (const float*)d_in[6];
    const float* Tp = (const float*)d_in[7];
    float* out = (float*)d_out;

    spectral_coherence_kernel<<<dim3(NK1), dim3(512), 0, stream>>>(
        k1, dy, dz, k2, k3, Lp, Mp, Tp, out);
}